// GCN_LSTM_Model_33741263077893
// MI455X (gfx1250) — compile-verified
//
#include <hip/hip_runtime.h>
#include <math.h>

#define N_NODES 50000
#define N_EDGES 800000
#define NB 2
#define NT 8
#define F_IN 32
#define H_GCN 64
#define H_LSTM 128
#define F_OUT 32

typedef __attribute__((ext_vector_type(2))) float v2f;
typedef __attribute__((ext_vector_type(8))) float v8f;

// ---------------------------------------------------------------------------
// Kernel 1: init deg[n]=1 (self loop), y=0
// ---------------------------------------------------------------------------
__global__ void init_kernel(float* __restrict__ deg, float* __restrict__ y) {
  int i = blockIdx.x * blockDim.x + threadIdx.x;
  if (i < N_NODES) deg[i] = 1.0f;
  if (i < NB * NT * F_IN) y[i] = 0.0f;
}

// ---------------------------------------------------------------------------
// Kernel 2: deg[col[e]] += 1
// ---------------------------------------------------------------------------
__global__ void deg_kernel(const int* __restrict__ col, float* __restrict__ deg) {
  int e = blockIdx.x * blockDim.x + threadIdx.x;
  if (e < N_EDGES) atomicAdd(&deg[col[e]], 1.0f);
}

// ---------------------------------------------------------------------------
// Kernel 3: inv0 = rsqrt(deg); c0 = 1/deg (self-loop term of batch-0 coeff)
// ---------------------------------------------------------------------------
__global__ void inv_kernel(const float* __restrict__ deg, float* __restrict__ inv0,
                           float* __restrict__ c0) {
  int i = blockIdx.x * blockDim.x + threadIdx.x;
  if (i < N_NODES) {
    float d = deg[i];
    inv0[i] = rsqrtf(d);
    c0[i] = 1.0f / d;
  }
}

// ---------------------------------------------------------------------------
// Kernel 4: c0[row[e]] += inv0[row[e]] * inv0[col[e]]   (aggregation weight)
// ---------------------------------------------------------------------------
__global__ void wsum_kernel(const int* __restrict__ row, const int* __restrict__ col,
                            const float* __restrict__ inv0, float* __restrict__ c0) {
  int e = blockIdx.x * blockDim.x + threadIdx.x;
  if (e < N_EDGES) {
    int r = row[e];
    atomicAdd(&c0[r], inv0[r] * inv0[col[e]]);
  }
}

// ---------------------------------------------------------------------------
// Kernel 5 (the HBM-bound one): y[bt][f] = sum_n c[b,n] * x[b,t,n,f]
// lane = feature (coalesced 128B/wave), one node per wave per iteration.
// ---------------------------------------------------------------------------
__global__ void node_reduce_kernel(const float* __restrict__ x,
                                   const float* __restrict__ c0,
                                   float* __restrict__ y) {
  const int bt   = blockIdx.y;          // b*NT + t, 0..15
  const int lane = threadIdx.x & 31;    // feature
  const int wave = threadIdx.x >> 5;    // 0..7
  const int b    = bt >> 3;             // NT == 8
  const float* __restrict__ xbt = x + (size_t)bt * N_NODES * F_IN;

  float acc = 0.0f;
  for (int n = blockIdx.x * 8 + wave; n < N_NODES; n += gridDim.x * 8) {
    float coef = (b == 0) ? c0[n] : 1.0f;
    acc = fmaf(coef, xbt[(size_t)n * F_IN + lane], acc);
  }

  __shared__ float part[8][32];
  part[wave][lane] = acc;
  __syncthreads();
  if (threadIdx.x < 32) {
    float s = 0.0f;
#pragma unroll
    for (int w = 0; w < 8; ++w) s += part[w][threadIdx.x];
    atomicAdd(&y[bt * F_IN + threadIdx.x], s);
  }
}

// ---------------------------------------------------------------------------
// Kernel 6: hseq(16x64) = (y(16x32) @ W_gcn(32x64)) / N + b_gcn
// Single block, 4 waves; each wave owns one 16-wide N tile.
// V_WMMA_F32_16X16X4_F32, K stepped 8x4. f32 A/B fragment layout per ISA
// 7.12.2: lanes 0-15 hold K = k,k+1 in the 2 VGPRs; lanes 16-31 hold K+2,K+3.
// ---------------------------------------------------------------------------
__global__ void hseq_wmma_kernel(const float* __restrict__ y,
                                 const float* __restrict__ Wg,
                                 const float* __restrict__ bg,
                                 float* __restrict__ hseq) {
  const int lane = threadIdx.x & 31;
  const int wv   = threadIdx.x >> 5;        // N tile 0..3
  const int m    = lane & 15;               // A row / D col-within-tile
  const int koff = (lane >> 4) * 2;         // half-wave K offset
  const int ncol = wv * 16 + (lane & 15);   // output column 0..63

  v8f acc = {};
  for (int k = 0; k < F_IN; k += 4) {
    v2f a, bf;
    a.x  = y[m * F_IN + (k + koff)];
    a.y  = y[m * F_IN + (k + koff + 1)];
    bf.x = Wg[(k + koff) * H_GCN + ncol];
    bf.y = Wg[(k + koff + 1) * H_GCN + ncol];
    acc = __builtin_amdgcn_wmma_f32_16x16x4_f32(
        /*neg_a=*/false, a, /*neg_b=*/false, bf,
        /*c_mod=*/(short)0, acc, /*reuse_a=*/false, /*reuse_b=*/false);
  }

  const float invN = 1.0f / (float)N_NODES;
  const float bias = bg[ncol];
#pragma unroll
  for (int j = 0; j < 8; ++j) {
    int rowm = j + ((lane >> 4) << 3);      // VGPR j: M=j (lo half), M=j+8 (hi)
    hseq[rowm * H_GCN + ncol] = acc[j] * invN + bias;
  }
}

// ---------------------------------------------------------------------------
// Kernel 7: LSTM over T=8 + final FC. Single block, tiny, latency-bound.
// ---------------------------------------------------------------------------
__device__ __forceinline__ float sigm(float x) { return 1.0f / (1.0f + expf(-x)); }

__global__ void lstm_fc_kernel(const float* __restrict__ hseq,
                               const float* __restrict__ W_ih,
                               const float* __restrict__ W_hh,
                               const float* __restrict__ b_ih,
                               const float* __restrict__ b_hh,
                               const float* __restrict__ W_fc,
                               const float* __restrict__ b_fc,
                               float* __restrict__ out) {
  __shared__ float hs[NB][H_LSTM];
  __shared__ float cs[NB][H_LSTM];
  __shared__ float g[NB][4 * H_LSTM];
  const int tid = threadIdx.x;

  if (tid < NB * H_LSTM) {
    hs[tid >> 7][tid & 127] = 0.0f;
    cs[tid >> 7][tid & 127] = 0.0f;
  }
  __syncthreads();

  for (int t = 0; t < NT; ++t) {
    // gates: 2 x 512 outputs, each = b_ih+b_hh + x.W_ih^T + h.W_hh^T
    for (int jg = tid; jg < NB * 4 * H_LSTM; jg += blockDim.x) {
      int b = jg >> 9;
      int j = jg & 511;
      float acc = b_ih[j] + b_hh[j];
      const float* __restrict__ xr = hseq + (b * NT + t) * H_GCN;
      const float* __restrict__ wi = W_ih + j * H_GCN;
#pragma unroll 4
      for (int k = 0; k < H_GCN; ++k) acc = fmaf(xr[k], wi[k], acc);
      const float* __restrict__ wh = W_hh + j * H_LSTM;
#pragma unroll 4
      for (int k = 0; k < H_LSTM; ++k) acc = fmaf(hs[b][k], wh[k], acc);
      g[b][j] = acc;
    }
    __syncthreads();
    if (tid < NB * H_LSTM) {
      int b = tid >> 7, k = tid & 127;
      float ig = g[b][k];
      float fg = g[b][H_LSTM + k];
      float gg = g[b][2 * H_LSTM + k];
      float og = g[b][3 * H_LSTM + k];
      float c = sigm(fg) * cs[b][k] + sigm(ig) * tanhf(gg);
      cs[b][k] = c;
      hs[b][k] = sigm(og) * tanhf(c);
    }
    __syncthreads();
  }

  if (tid < NB * F_OUT) {
    int b = tid >> 5, o = tid & 31;
    float acc = b_fc[o];
    const float* __restrict__ wf = W_fc + o * H_LSTM;
#pragma unroll 4
    for (int k = 0; k < H_LSTM; ++k) acc = fmaf(hs[b][k], wf[k], acc);
    out[b * F_OUT + o] = acc;
  }
}

// ---------------------------------------------------------------------------
extern "C" void kernel_launch(void* const* d_in, const int* in_sizes, int n_in,
                              void* d_out, int out_size, void* d_ws, size_t ws_size,
                              hipStream_t stream) {
  const float* x    = (const float*)d_in[0];   // (B,T,N,F_IN)
  const int*   ei   = (const int*)d_in[1];     // (2,E): row then col
  const float* Wg   = (const float*)d_in[2];   // (F_IN,H_GCN)
  const float* bg   = (const float*)d_in[3];   // (H_GCN)
  const float* Wih  = (const float*)d_in[4];   // (4H,H_GCN)
  const float* Whh  = (const float*)d_in[5];   // (4H,H_LSTM)
  const float* bih  = (const float*)d_in[6];
  const float* bhh  = (const float*)d_in[7];
  const float* Wfc  = (const float*)d_in[8];   // (F_OUT,H_LSTM)
  const float* bfc  = (const float*)d_in[9];
  float* out = (float*)d_out;

  float* ws   = (float*)d_ws;
  float* deg  = ws;                    // N
  float* inv0 = ws + N_NODES;          // N
  float* c0   = ws + 2 * N_NODES;      // N
  float* y    = ws + 3 * N_NODES;      // 16*32
  float* hseq = y + NB * NT * F_IN;    // 16*64

  const int* row = ei;
  const int* col = ei + N_EDGES;

  const int nodeBlocks = (N_NODES + 255) / 256;
  const int edgeBlocks = (N_EDGES + 255) / 256;

  init_kernel<<<nodeBlocks, 256, 0, stream>>>(deg, y);
  deg_kernel<<<edgeBlocks, 256, 0, stream>>>(col, deg);
  inv_kernel<<<nodeBlocks, 256, 0, stream>>>(deg, inv0, c0);
  wsum_kernel<<<edgeBlocks, 256, 0, stream>>>(row, col, inv0, c0);

  dim3 rgrid(64, NB * NT);             // 1024 blocks; HBM-bound streaming pass
  node_reduce_kernel<<<rgrid, 256, 0, stream>>>(x, c0, y);

  hseq_wmma_kernel<<<1, 128, 0, stream>>>(y, Wg, bg, hseq);
  lstm_fc_kernel<<<1, 256, 0, stream>>>(hseq, Wih, Whh, bih, bhh, Wfc, bfc, out);
}